// SMPL_3839700763008
// MI455X (gfx1250) — compile-verified
//
#include <hip/hip_runtime.h>
#include <hip/hip_bf16.h>
#include <math.h>

#define B_   512
#define NV_  6890
#define NJ_  24
#define NB_  10
#define NP_  207
#define KC_  220   // combined K: 207 lrotmin + 10 beta + 1 (v_template) + 2 pad
#define KMAIN_ 204 // branch-free posedirs-only K range (51 steps of 4)

typedef float v2f __attribute__((ext_vector_type(2)));
typedef float v4f __attribute__((ext_vector_type(4)));
typedef float v8f __attribute__((ext_vector_type(8)));

__constant__ int PARENTC[23] = {0,0,0,1,2,3,4,5,6,7,8,9,9,9,12,13,14,16,17,18,19,20,21};

__device__ __forceinline__ float acomb(const float* __restrict__ pd,
                                       const float* __restrict__ sd,
                                       const float* __restrict__ vt,
                                       int v, int d, int k) {
  if (k < NP_)             return pd[(size_t)v * (3 * NP_) + d * NP_ + k];
  else if (k < NP_ + NB_)  return sd[(size_t)v * (3 * NB_) + d * NB_ + (k - NP_)];
  else if (k == NP_ + NB_) return vt[v * 3 + d];
  else                     return 0.0f;
}

// ---------------------------------------------------------------------------
// Kernel 0: pack the segment-crossing K-tail [204..219] of the combined
// A-matrix into a contiguous (NV,3,16) block so the fused GEMM loop is
// branch-free.
// ---------------------------------------------------------------------------
__global__ void smpl_atail_kernel(const float* __restrict__ pd,
                                  const float* __restrict__ sd,
                                  const float* __restrict__ vt,
                                  float* __restrict__ At) {
  const int t = blockIdx.x * 256 + threadIdx.x;
  if (t >= NV_ * 3 * 16) return;
  const int kk = t & 15;
  const int vd = t >> 4;
  At[t] = acomb(pd, sd, vt, vd / 3, vd % 3, KMAIN_ + kk);
}

// ---------------------------------------------------------------------------
// Kernel 1: JS[j,d,k] = sum_v Jreg[j,v]*shapedirs[v,d,k];  Jt = Jreg@v_template
// grid (24, 33), block 256
// ---------------------------------------------------------------------------
__global__ void smpl_jreg_kernel(const float* __restrict__ Jreg,
                                 const float* __restrict__ shapedirs,
                                 const float* __restrict__ v_template,
                                 float* __restrict__ JS, float* __restrict__ Jt) {
  __shared__ float red[256];
  const int j = blockIdx.x;
  const int o = blockIdx.y;              // 0..29 -> JS, 30..32 -> Jt
  const float* jr = Jreg + (size_t)j * NV_;
  float acc = 0.f;
  if (o < 30) {
    const int d = o / NB_, k = o % NB_;
    for (int v = threadIdx.x; v < NV_; v += 256)
      acc += jr[v] * shapedirs[(size_t)v * (3 * NB_) + d * NB_ + k];
  } else {
    const int d = o - 30;
    for (int v = threadIdx.x; v < NV_; v += 256)
      acc += jr[v] * v_template[v * 3 + d];
  }
  red[threadIdx.x] = acc;
  __syncthreads();
  for (int s = 128; s > 0; s >>= 1) {
    if (threadIdx.x < s) red[threadIdx.x] += red[threadIdx.x + s];
    __syncthreads();
  }
  if (threadIdx.x == 0) {
    if (o < 30) JS[(j * 3 + o / NB_) * NB_ + (o % NB_)] = red[0];
    else        Jt[j * 3 + (o - 30)] = red[0];
  }
}

// ---------------------------------------------------------------------------
// Kernel 2: rodrigues -> R (ws) and combined K-vector c[b] (ws)
// ---------------------------------------------------------------------------
__global__ void smpl_rot_kernel(const float* __restrict__ pose,
                                const float* __restrict__ beta,
                                float* __restrict__ Rws, float* __restrict__ cws) {
  const int t = blockIdx.x * blockDim.x + threadIdx.x;
  if (t >= B_ * NJ_) return;
  const int b = t / NJ_, j = t % NJ_;
  const float tx = pose[b * (NJ_ * 3) + j * 3 + 0];
  const float ty = pose[b * (NJ_ * 3) + j * 3 + 1];
  const float tz = pose[b * (NJ_ * 3) + j * 3 + 2];
  const float ex = tx + 1e-8f, ey = ty + 1e-8f, ez = tz + 1e-8f;
  const float angle = sqrtf(ex * ex + ey * ey + ez * ez);
  const float inv = 1.0f / angle;
  const float nx = tx * inv, ny = ty * inv, nz = tz * inv;
  const float hf = 0.5f * angle;
  float w = cosf(hf), s = sinf(hf);
  float qx = s * nx, qy = s * ny, qz = s * nz;
  const float qn = 1.0f / sqrtf(w * w + qx * qx + qy * qy + qz * qz);
  w *= qn; qx *= qn; qy *= qn; qz *= qn;
  const float w2 = w * w, x2 = qx * qx, y2 = qy * qy, z2 = qz * qz;
  const float wx = w * qx, wy = w * qy, wz = w * qz;
  const float xy = qx * qy, xz = qx * qz, yz = qy * qz;
  const float r00 = w2 + x2 - y2 - z2, r01 = 2.f*xy - 2.f*wz, r02 = 2.f*wy + 2.f*xz;
  const float r10 = 2.f*wz + 2.f*xy, r11 = w2 - x2 + y2 - z2, r12 = 2.f*yz - 2.f*wx;
  const float r20 = 2.f*xz - 2.f*wy, r21 = 2.f*wx + 2.f*yz, r22 = w2 - x2 - y2 + z2;
  float* R = Rws + (size_t)(b * NJ_ + j) * 9;
  R[0] = r00; R[1] = r01; R[2] = r02;
  R[3] = r10; R[4] = r11; R[5] = r12;
  R[6] = r20; R[7] = r21; R[8] = r22;
  if (j >= 1) {
    float* c = cws + (size_t)b * KC_ + (j - 1) * 9;
    c[0] = r00 - 1.f; c[1] = r01;       c[2] = r02;
    c[3] = r10;       c[4] = r11 - 1.f; c[5] = r12;
    c[6] = r20;       c[7] = r21;       c[8] = r22 - 1.f;
  } else {
    float* c = cws + (size_t)b * KC_ + NP_;
    for (int k = 0; k < NB_; ++k) c[k] = beta[b * NB_ + k];
    c[NB_] = 1.f; c[NB_ + 1] = 0.f; c[NB_ + 2] = 0.f;
  }
}

// ---------------------------------------------------------------------------
// Kernel 3: kinematic chain -> corrected G' (b, 24, 3x4). One thread per batch.
// ---------------------------------------------------------------------------
__global__ void smpl_chain_kernel(const float* __restrict__ JS, const float* __restrict__ Jt,
                                  const float* __restrict__ beta, const float* __restrict__ Rws,
                                  float* __restrict__ Gpo) {
  __shared__ float Gsh[16][NJ_ * 12];
  __shared__ float Jsh[16][NJ_ * 3];
  const int tid = threadIdx.x;
  const int b = blockIdx.x * 16 + tid;
  const float* bv = beta + b * NB_;
  float* J = Jsh[tid];
  for (int e = 0; e < NJ_ * 3; ++e) {
    float a = Jt[e];
    const float* js = JS + e * NB_;
    for (int k = 0; k < NB_; ++k) a += js[k] * bv[k];
    J[e] = a;
  }
  float* G = Gsh[tid];
  const float* Rb = Rws + (size_t)b * NJ_ * 9;
  for (int j = 0; j < NJ_; ++j) {
    const float* R = Rb + j * 9;
    float L[12];
    L[0] = R[0]; L[1] = R[1]; L[2]  = R[2];
    L[4] = R[3]; L[5] = R[4]; L[6]  = R[5];
    L[8] = R[6]; L[9] = R[7]; L[10] = R[8];
    if (j == 0) {
      L[3] = J[0]; L[7] = J[1]; L[11] = J[2];
      for (int i = 0; i < 12; ++i) G[i] = L[i];
    } else {
      const int p = PARENTC[j - 1];
      L[3]  = J[j * 3 + 0] - J[p * 3 + 0];
      L[7]  = J[j * 3 + 1] - J[p * 3 + 1];
      L[11] = J[j * 3 + 2] - J[p * 3 + 2];
      const float* gp = G + p * 12;
      float* go = G + j * 12;
      for (int m = 0; m < 3; ++m)
        for (int n = 0; n < 4; ++n) {
          float a = gp[m * 4 + 0] * L[n] + gp[m * 4 + 1] * L[4 + n] + gp[m * 4 + 2] * L[8 + n];
          if (n == 3) a += gp[m * 4 + 3];
          go[m * 4 + n] = a;
        }
    }
  }
  float* outb = Gpo + (size_t)b * NJ_ * 12;
  for (int j = 0; j < NJ_; ++j) {
    const float jx = J[j * 3 + 0], jy = J[j * 3 + 1], jz = J[j * 3 + 2];
    for (int m = 0; m < 3; ++m) {
      const float g0 = G[j * 12 + m * 4 + 0], g1 = G[j * 12 + m * 4 + 1];
      const float g2 = G[j * 12 + m * 4 + 2], g3 = G[j * 12 + m * 4 + 3];
      outb[j * 12 + m * 4 + 0] = g0;
      outb[j * 12 + m * 4 + 1] = g1;
      outb[j * 12 + m * 4 + 2] = g2;
      outb[j * 12 + m * 4 + 3] = g3 - (g0 * jx + g1 * jy + g2 * jz);
    }
  }
}

// ---------------------------------------------------------------------------
// Kernel 4: fused pose/shape blend GEMM + skinning GEMM, all WMMA f32.
// One wave per 16-vertex x 16-batch tile. Branch-free K loop.
// ---------------------------------------------------------------------------
__global__ void __launch_bounds__(32)
smpl_fused_kernel(const float* __restrict__ posedirs,
                  const float* __restrict__ Atail,
                  const float* __restrict__ weights,
                  const float* __restrict__ cws,
                  const float* __restrict__ Gp,
                  float* __restrict__ out) {
  __shared__ __align__(16) float GpL[16 * NJ_ * 12];   // 16 batches x 24 x 3x4
  const int lane = threadIdx.x;
  const int v0 = blockIdx.x * 16;
  const int b0 = blockIdx.y * 16;

  // stage G' tile (contiguous 4608 floats, 16B aligned) into LDS as b128
  const float* gsrc = Gp + (size_t)b0 * (NJ_ * 12);
  for (int i = lane * 4; i < 16 * NJ_ * 12; i += 32 * 4)
    *(v4f*)&GpL[i] = *(const v4f*)&gsrc[i];
  __syncthreads();

  const int hi   = lane >> 4;       // half-wave select
  const int lid  = lane & 15;
  const int koff = hi * 2;          // A/B fragment K sub-offset per ISA layout
  int vrow = v0 + lid; if (vrow >= NV_) vrow = NV_ - 1;   // A-frag row (M)
  const int bcol = lid;                                    // B-frag col (N)

  // per-lane base pointers (koff folded in)
  const float* pd0 = posedirs + (size_t)vrow * (3 * NP_) + koff;        // d stride NP_
  const float* at0 = Atail + (size_t)vrow * 48 + koff;                  // d stride 16
  const float* cr0 = cws + (size_t)(b0 + bcol) * KC_ + koff;

  v8f acc0 = {0,0,0,0,0,0,0,0};
  v8f acc1 = {0,0,0,0,0,0,0,0};
  v8f acc2 = {0,0,0,0,0,0,0,0};

  // phase 1a: 51 branch-free steps over posedirs (k = 0..203)
  #pragma unroll 4
  for (int s = 0; s < KMAIN_ / 4; ++s) {
    const int k = s * 4;
    v2f a0 = { pd0[k],           pd0[k + 1] };
    v2f a1 = { pd0[NP_ + k],     pd0[NP_ + k + 1] };
    v2f a2 = { pd0[2 * NP_ + k], pd0[2 * NP_ + k + 1] };
    v2f bf = { cr0[k],           cr0[k + 1] };
    acc0 = __builtin_amdgcn_wmma_f32_16x16x4_f32(false, a0, false, bf, (short)0, acc0, false, false);
    acc1 = __builtin_amdgcn_wmma_f32_16x16x4_f32(false, a1, false, bf, (short)0, acc1, false, false);
    acc2 = __builtin_amdgcn_wmma_f32_16x16x4_f32(false, a2, false, bf, (short)0, acc2, false, false);
  }
  // phase 1b: 4 branch-free tail steps from packed Atail (k = 204..219)
  #pragma unroll
  for (int s = 0; s < 4; ++s) {
    const int kk = s * 4;
    v2f a0 = { at0[kk],      at0[kk + 1] };
    v2f a1 = { at0[16 + kk], at0[16 + kk + 1] };
    v2f a2 = { at0[32 + kk], at0[32 + kk + 1] };
    v2f bf = { cr0[KMAIN_ + kk], cr0[KMAIN_ + kk + 1] };
    acc0 = __builtin_amdgcn_wmma_f32_16x16x4_f32(false, a0, false, bf, (short)0, acc0, false, false);
    acc1 = __builtin_amdgcn_wmma_f32_16x16x4_f32(false, a1, false, bf, (short)0, acc1, false, false);
    acc2 = __builtin_amdgcn_wmma_f32_16x16x4_f32(false, a2, false, bf, (short)0, acc2, false, false);
  }

  // phase 2: skinning T = weights x G' as WMMA (K=24), fragment-aligned with p
  v2f aw[6];
  #pragma unroll
  for (int s = 0; s < 6; ++s) {
    const int j = s * 4 + koff;
    aw[s].x = weights[(size_t)vrow * NJ_ + j];
    aw[s].y = weights[(size_t)vrow * NJ_ + j + 1];
  }
  #pragma unroll
  for (int m = 0; m < 3; ++m) {
    v8f t0 = {0,0,0,0,0,0,0,0};
    v8f t1 = {0,0,0,0,0,0,0,0};
    v8f t2 = {0,0,0,0,0,0,0,0};
    v8f t3 = {0,0,0,0,0,0,0,0};
    #pragma unroll
    for (int s = 0; s < 6; ++s) {
      const int j = s * 4 + koff;
      const float* g0 = &GpL[bcol * (NJ_ * 12) + j * 12 + m * 4];
      const float* g1 = &GpL[bcol * (NJ_ * 12) + (j + 1) * 12 + m * 4];
      v2f bfrag;
      bfrag.x = g0[0]; bfrag.y = g1[0];
      t0 = __builtin_amdgcn_wmma_f32_16x16x4_f32(false, aw[s], false, bfrag, (short)0, t0, false, false);
      bfrag.x = g0[1]; bfrag.y = g1[1];
      t1 = __builtin_amdgcn_wmma_f32_16x16x4_f32(false, aw[s], false, bfrag, (short)0, t1, false, false);
      bfrag.x = g0[2]; bfrag.y = g1[2];
      t2 = __builtin_amdgcn_wmma_f32_16x16x4_f32(false, aw[s], false, bfrag, (short)0, t2, false, false);
      bfrag.x = g0[3]; bfrag.y = g1[3];
      t3 = __builtin_amdgcn_wmma_f32_16x16x4_f32(false, aw[s], false, bfrag, (short)0, t3, false, false);
    }
    // v_m = T[m,0]*p0 + T[m,1]*p1 + T[m,2]*p2 + T[m,3]  (layouts identical)
    v8f o = t0 * acc0 + t1 * acc1 + t2 * acc2 + t3;
    #pragma unroll
    for (int r = 0; r < 8; ++r) {
      const int vr = v0 + r + hi * 8;
      if (vr < NV_) out[((size_t)(b0 + bcol) * NV_ + vr) * 3 + m] = o[r];
    }
  }
}

// ---------------------------------------------------------------------------
// Workspace layout (floats):
//   JS  @ 0       (720)
//   Jt  @ 768     (72)
//   R   @ 1024    (512*24*9  = 110592)
//   c   @ 111616  (512*220   = 112640)
//   G'  @ 224256  (512*24*12 = 147456)
//   At  @ 371712  (6890*3*16 = 330720)   total 702432 floats = 2.81 MB
// ---------------------------------------------------------------------------
extern "C" void kernel_launch(void* const* d_in, const int* in_sizes, int n_in,
                              void* d_out, int out_size, void* d_ws, size_t ws_size,
                              hipStream_t stream) {
  const float* pose       = (const float*)d_in[0];
  const float* beta       = (const float*)d_in[1];
  const float* v_template = (const float*)d_in[2];
  const float* shapedirs  = (const float*)d_in[3];
  const float* posedirs   = (const float*)d_in[4];
  const float* Jreg       = (const float*)d_in[5];
  const float* weights    = (const float*)d_in[6];
  float* ws = (float*)d_ws;
  float* JS = ws;
  float* Jt = ws + 768;
  float* Rw = ws + 1024;
  float* cw = ws + 111616;
  float* Gp = ws + 224256;
  float* At = ws + 371712;
  float* out = (float*)d_out;

  smpl_atail_kernel<<<(NV_ * 3 * 16 + 255) / 256, 256, 0, stream>>>(
      posedirs, shapedirs, v_template, At);
  smpl_jreg_kernel<<<dim3(24, 33), 256, 0, stream>>>(Jreg, shapedirs, v_template, JS, Jt);
  smpl_rot_kernel<<<(B_ * NJ_ + 255) / 256, 256, 0, stream>>>(pose, beta, Rw, cw);
  smpl_chain_kernel<<<B_ / 16, 16, 0, stream>>>(JS, Jt, beta, Rw, Gp);
  smpl_fused_kernel<<<dim3((NV_ + 15) / 16, B_ / 16), 32, 0, stream>>>(
      posedirs, At, weights, cw, Gp, out);
}